// shuffleAug_89730456748427
// MI455X (gfx1250) — compile-verified
//
#include <hip/hip_runtime.h>
#include <stdint.h>

// ---------------------------------------------------------------------------
// shuffleAug: per-sample D4 spatial transform of two (B,C,128,128) f32 tensors.
// The 5 sequential conditional gathers in the reference compose into a single
// per-sample mapping (s = f2):
//   s==0: out[i,j] = in[(f0^f3)?127-i:i, (f1^f4)?127-j:j]
//   s==1: out[i,j] = in[(f0^f4)?127-j:j, (f1^f3)?127-i:i]
// which is affine in (i,j):  src_lds_addr = base + i*ci + j*cj  (block-uniform
// base/ci/cj). Pure data movement: 512 MiB traffic -> ~22us at 23.3 TB/s.
// One block per 64KB plane: CDNA5 async-stage into padded LDS
// (global_load_async_to_lds_b128, ASYNCcnt), then coalesced float4 stores.
// ---------------------------------------------------------------------------

#define HH    128
#define PITCH 132   // 132*4 = 528 B row pitch: 16B-aligned, 132%64=4 -> <=2-way bank conflicts

#if defined(__HIP_DEVICE_COMPILE__)
#  if __has_builtin(__builtin_amdgcn_global_load_async_to_lds_b128)
#    define USE_ASYNC 1
#  else
#    define USE_ASYNC 0
#    warning "DEVICE PASS: __builtin_amdgcn_global_load_async_to_lds_b128 NOT available - using VGPR staging fallback"
#  endif
#else
#  define USE_ASYNC 0
#endif

// Builtin signature (from clang diagnostic): param 0 is AS1 int4*, param 1 AS3.
typedef int v4i __attribute__((ext_vector_type(4)));
typedef __attribute__((address_space(1))) v4i as1_v4i;
typedef __attribute__((address_space(3))) v4i as3_v4i;

__device__ __forceinline__ void stage16(const float* __restrict__ g, float* l) {
#if USE_ASYNC
  // Flat LDS addresses carry the wave-relative LDS offset in the low 32 bits
  // (aperture in the high word), so truncation yields a valid AS3 pointer.
  __builtin_amdgcn_global_load_async_to_lds_b128(
      (as1_v4i*)(uintptr_t)g,
      (as3_v4i*)(uint32_t)(uintptr_t)l,
      /*imm offset=*/0, /*cpol=*/0);
#else
  *reinterpret_cast<float4*>(l) = *reinterpret_cast<const float4*>(g);
#endif
}

__device__ __forceinline__ void wait_async_done() {
#if USE_ASYNC
#  if __has_builtin(__builtin_amdgcn_s_wait_asynccnt)
  __builtin_amdgcn_s_wait_asynccnt(0);
#  else
  asm volatile("s_wait_asynccnt 0" ::: "memory");
#  endif
#endif
}

__global__ __launch_bounds__(256) void shuffleAug_kernel(
    const float* __restrict__ x1, const float* __restrict__ x2,
    const int* __restrict__ flip, float* __restrict__ out, int B, int C) {
  __shared__ float tile[HH * PITCH];

  const int planes = B * C;                 // planes per tensor
  const int p   = blockIdx.x;               // [0, 2*planes)
  const int t   = (p >= planes) ? 1 : 0;    // which input tensor
  const int rem = p - t * planes;           // (b,c) flat within tensor
  const int b   = rem / C;

  const float* __restrict__ src = (t ? x2 : x1) + (size_t)rem * (HH * HH);
  float* __restrict__ dst       = out + (size_t)p * (HH * HH);  // outputs concatenated in order

  // flip_bits is (5, B) int32, values 0/1
  const int f0 = flip[0 * B + b] & 1;
  const int f1 = flip[1 * B + b] & 1;
  const int f2 = flip[2 * B + b] & 1;
  const int f3 = flip[3 * B + b] & 1;
  const int f4 = flip[4 * B + b] & 1;
  const int s  = f2;
  const int rf = s ? (f0 ^ f4) : (f0 ^ f3);
  const int cf = s ? (f1 ^ f3) : (f1 ^ f4);

  // Affine source-address map (block-uniform coefficients):
  //   lds_addr(i,j) = base + i*ci + j*cj
  const int base = (rf ? (HH - 1) * PITCH : 0) + (cf ? (HH - 1) : 0);
  const int ci   = s ? (cf ? -1 : 1) : (rf ? -PITCH : PITCH);
  const int cj   = s ? (rf ? -PITCH : PITCH) : (cf ? -1 : 1);

  const int tid = threadIdx.x;              // 256 threads = 8 waves (wave32)

  // ---- Stage 1: plane -> LDS (padded pitch), 16B granules, fully coalesced
#pragma unroll
  for (int k = 0; k < 16; ++k) {
    const int idx4 = (k << 8) + tid;        // 4096 float4 per plane
    const int r    = idx4 >> 5;             // 32 float4 per 128-float row
    const int c0   = (idx4 & 31) << 2;
    stage16(src + r * HH + c0, &tile[r * PITCH + c0]);
  }
  wait_async_done();
  __syncthreads();

  // ---- Stage 2: permuted LDS reads -> coalesced float4 global stores
  const int i0   = tid >> 5;                // this thread's first output row
  const int j0   = (tid & 31) << 2;         // this thread's output col group
  int addr0      = base + i0 * ci + j0 * cj;
#pragma unroll
  for (int k = 0; k < 16; ++k) {            // 16 output rows per thread, stride 8
    const int i = i0 + (k << 3);
    float4 v;
    float* vp = reinterpret_cast<float*>(&v);
    int a = addr0 + (k << 3) * ci;
#pragma unroll
    for (int e = 0; e < 4; ++e) {
      vp[e] = tile[a];
      a += cj;
    }
    *reinterpret_cast<float4*>(dst + i * HH + j0) = v;
  }
}

extern "C" void kernel_launch(void* const* d_in, const int* in_sizes, int n_in,
                              void* d_out, int out_size, void* d_ws, size_t ws_size,
                              hipStream_t stream) {
  const float* x1  = (const float*)d_in[0];
  const float* x2  = (const float*)d_in[1];
  const int* flip  = (const int*)d_in[2];
  float* out       = (float*)d_out;

  const int B = in_sizes[2] / 5;                    // flip_bits is (5, B)
  const int C = in_sizes[0] / (B * HH * HH);        // x1 is (B, C, 128, 128)
  const int nBlocks = 2 * B * C;                    // one block per 128x128 plane

  shuffleAug_kernel<<<dim3(nBlocks), dim3(256), 0, stream>>>(x1, x2, flip, out, B, C);
}